// FapeLoss_54039278518369
// MI455X (gfx1250) — compile-verified
//
#include <hip/hip_runtime.h>
#include <hip/hip_bf16.h>

typedef float v2f __attribute__((ext_vector_type(2)));
typedef float v8f __attribute__((ext_vector_type(8)));

#define FAPE_EPS 1e-8f
#define FAPE_CLAMP 10.0f
#define FAPE_INV_LS 0.1f

// ---------------------------------------------------------------------------
// Kernel 0: zero the per-batch accumulators (acc[B], S[B]) in workspace.
// ---------------------------------------------------------------------------
__global__ void fape_zero_kernel(float* acc, float* S, int B) {
    int t = threadIdx.x;
    if (t < B) { acc[t] = 0.0f; S[t] = 0.0f; }
}

// ---------------------------------------------------------------------------
// Kernel 1: u = pred - true (padded to float4), n2 = |u|^2, S[b] = sum(mask).
// ---------------------------------------------------------------------------
__global__ void fape_prep_kernel(const float* __restrict__ pred,
                                 const float* __restrict__ tru,
                                 const float* __restrict__ mask,
                                 float4* __restrict__ u4,
                                 float* __restrict__ n2,
                                 float* __restrict__ S,
                                 int BN, int N) {
    int idx = blockIdx.x * blockDim.x + threadIdx.x;
    if (idx >= BN) return;
    float ux = pred[3 * idx + 0] - tru[3 * idx + 0];
    float uy = pred[3 * idx + 1] - tru[3 * idx + 1];
    float uz = pred[3 * idx + 2] - tru[3 * idx + 2];
    u4[idx] = make_float4(ux, uy, uz, 0.0f);
    n2[idx] = ux * ux + uy * uy + uz * uz;
    atomicAdd(&S[idx / N], mask[idx]);
}

// ---------------------------------------------------------------------------
// Kernel 2: all-pairs tile sweep with V_WMMA_F32_16X16X4_F32.
// Grid: B * (N/16) blocks of 256 threads (8 waves). Block (b, i_tile);
// each wave strides over j-tiles. Gram tile G = A(16x4) x B(4x16), then
// err = clamp(sqrt(n_i + n_j - 2G + eps), 0, 10)/10 * m_i * m_j.
// Per-row accumulators defer the mi[r] and 1/length_scale factors out of
// the hot loop; EPS is folded into the precomputed row norms.
// ---------------------------------------------------------------------------
__global__ __launch_bounds__(256) void fape_pair_kernel(
    const float4* __restrict__ u4,
    const float* __restrict__ n2,
    const float* __restrict__ mask,
    float* __restrict__ acc,
    int N, int nTiles) {
    const int b      = blockIdx.x / nTiles;
    const int i_tile = blockIdx.x % nTiles;
    const int lane   = threadIdx.x & 31;
    // wave id is uniform within a wave: force scalar loop control
    const int wave   = __builtin_amdgcn_readfirstlane(threadIdx.x >> 5);
    const int half   = lane >> 4;   // 0: lanes 0-15, 1: lanes 16-31
    const int l15    = lane & 15;

    const float4* __restrict__ Ub  = u4   + (size_t)b * N;
    const float*  __restrict__ n2b = n2   + (size_t)b * N;
    const float*  __restrict__ mb  = mask + (size_t)b * N;

    // --- A fragment: 16x4 f32, lanes 0-15 hold (K=0,K=1), lanes 16-31 (K=2,K=3)
    const int i_base = i_tile * 16;
    float4 ua = Ub[i_base + l15];
    v2f A;
    A.x = half ? ua.z : ua.x;   // K = 0 or 2
    A.y = half ? 0.0f : ua.y;   // K = 1 or 3 (K=3 is the zero pad column)

    // --- epilogue constants for rows: C/D VGPR r -> M = r + 8*half, N = l15
    float niE[8], mi[8];
#pragma unroll
    for (int r = 0; r < 8; ++r) {
        int i = i_base + r + half * 8;
        niE[r] = n2b[i] + FAPE_EPS;   // fold EPS: max(d2,0)+eps == max(d2+eps, eps)
        mi[r]  = mb[i];
    }

    float accr[8];
#pragma unroll
    for (int r = 0; r < 8; ++r) accr[r] = 0.0f;

#pragma unroll 2
    for (int jt = wave; jt < nTiles; jt += 8) {
        const int j_base = jt * 16;

        // --- B fragment: 4x16 f32, VGPR0 rows K=0 (lanes 0-15) / K=2 (16-31),
        //     VGPR1 rows K=1 / K=3; column N = l15.
        float4 ub = Ub[j_base + l15];
        v2f Bf;
        Bf.x = half ? ub.z : ub.x;
        Bf.y = half ? 0.0f : ub.y;

        float nj = n2b[j_base + l15];
        float mj = mb[j_base + l15];

        v8f c = {};
        c = __builtin_amdgcn_wmma_f32_16x16x4_f32(
            /*neg_a=*/false, A, /*neg_b=*/false, Bf,
            /*c_mod=*/(short)0, c, /*reuse_a=*/false, /*reuse_b=*/false);

#pragma unroll
        for (int r = 0; r < 8; ++r) {
            float t = niE[r] + nj - 2.0f * c[r];       // pk_fma-able
            t = fmaxf(t, FAPE_EPS);
            float e = __builtin_amdgcn_sqrtf(t);       // raw v_sqrt_f32 (TRANS)
            e = fminf(e, FAPE_CLAMP);
            accr[r] = fmaf(e, mj, accr[r]);
        }
    }

    // fold row mask and 1/length_scale after the hot loop
    float lacc = 0.0f;
#pragma unroll
    for (int r = 0; r < 8; ++r) lacc = fmaf(accr[r], mi[r], lacc);
    lacc *= FAPE_INV_LS;

    // --- block reduction, one atomic per block
    __shared__ float red[256];
    red[threadIdx.x] = lacc;
    __syncthreads();
#pragma unroll
    for (int s = 128; s > 0; s >>= 1) {
        if (threadIdx.x < s) red[threadIdx.x] += red[threadIdx.x + s];
        __syncthreads();
    }
    if (threadIdx.x == 0) atomicAdd(&acc[b], red[0]);
}

// ---------------------------------------------------------------------------
// Kernel 3: out[b] = acc[b] / (eps + S[b])^2
// ---------------------------------------------------------------------------
__global__ void fape_finalize_kernel(const float* __restrict__ acc,
                                     const float* __restrict__ S,
                                     float* __restrict__ out, int B) {
    int b = threadIdx.x;
    if (b < B) {
        float d = FAPE_EPS + S[b];
        out[b] = acc[b] / (d * d);
    }
}

extern "C" void kernel_launch(void* const* d_in, const int* in_sizes, int n_in,
                              void* d_out, int out_size, void* d_ws, size_t ws_size,
                              hipStream_t stream) {
    const float* pred = (const float*)d_in[0];
    const float* tru  = (const float*)d_in[1];
    const float* mask = (const float*)d_in[2];
    float* out = (float*)d_out;

    const int BN = in_sizes[2];      // B*N
    const int B  = out_size;         // one scalar per batch
    const int N  = BN / B;
    const int nTiles = N / 16;

    // Workspace layout (floats): [acc: B][S: B][pad to 64B][u4: BN*4][n2: BN]
    float* ws  = (float*)d_ws;
    float* acc = ws;
    float* S   = ws + B;
    float4* u4 = (float4*)(ws + 16);
    float*  n2 = (float*)(u4 + BN);

    fape_zero_kernel<<<1, 64, 0, stream>>>(acc, S, B);
    fape_prep_kernel<<<(BN + 255) / 256, 256, 0, stream>>>(pred, tru, mask, u4, n2, S, BN, N);
    fape_pair_kernel<<<B * nTiles, 256, 0, stream>>>(u4, n2, mask, acc, N, nTiles);
    fape_finalize_kernel<<<1, 64, 0, stream>>>(acc, S, out, B);
}